// GCN_57947698758286
// MI455X (gfx1250) — compile-verified
//
#include <hip/hip_runtime.h>

// GCN 2-layer forward for MI455X (gfx1250, wave32).
// - Dense GEMMs: V_WMMA_F32_16X16X4_F32 (fp32-exact vs reference), A staged in
//   LDS via GLOBAL_LOAD_ASYNC_TO_LDS_B128 (ASYNCcnt) — pure copy, since the
//   norm_src row scaling commutes with @W and is folded into the epilogue.
// - Edge aggregation: wave-per-edge float4 gathers + global_atomic_add_f32
//   (bandwidth-dominant phase: ~2.2 GB -> ~95us at 23.3 TB/s).

typedef __attribute__((ext_vector_type(2))) float v2f;
typedef __attribute__((ext_vector_type(8))) float v8f;

#define IN_F  256
#define H_F   128
#define N_CLS 40
#define N_PAD 48   // 3 x 16 WMMA column tiles for layer 2

// ---- CDNA5 async copy global->LDS (ASYNCcnt), per-lane 16B ----
__device__ __forceinline__ void async_g2l_b128(void* lds_ptr, const float* gaddr)
{
    unsigned lds_off = (unsigned)(unsigned long long)lds_ptr;  // low 32b of flat LDS ptr = as(3) offset
    asm volatile("global_load_async_to_lds_b128 %0, %1, off"
                 :: "v"(lds_off), "v"(gaddr)
                 : "memory");
}
__device__ __forceinline__ void wait_async0()
{
    asm volatile("s_wait_asynccnt 0x0" ::: "memory");
}

// ---------------- degrees + norms ----------------
__global__ void gcn_degree_kernel(const int* __restrict__ src, const int* __restrict__ dst,
                                  float* __restrict__ deg_s, float* __restrict__ deg_d, int E)
{
    int i = blockIdx.x * blockDim.x + threadIdx.x;
    int stride = gridDim.x * blockDim.x;
    for (; i < E; i += stride) {
        atomicAdd(&deg_s[src[i]], 1.0f);
        atomicAdd(&deg_d[dst[i]], 1.0f);
    }
}

__global__ void gcn_norm_kernel(float* __restrict__ ns, float* __restrict__ nd, int n)
{
    int i = blockIdx.x * blockDim.x + threadIdx.x;
    if (i < n) {
        ns[i] = 1.0f / sqrtf(fmaxf(ns[i], 1.0f));
        nd[i] = 1.0f / sqrtf(fmaxf(nd[i], 1.0f));
    }
}

// ---------------- W2 zero-pad 40 -> 48 cols ----------------
__global__ void gcn_padw2(const float* __restrict__ W2, float* __restrict__ Wp)
{
    int i = blockIdx.x * blockDim.x + threadIdx.x;
    if (i < H_F * N_PAD) {
        int k = i / N_PAD, c = i - k * N_PAD;
        Wp[i] = (c < N_CLS) ? W2[k * N_CLS + c] : 0.0f;
    }
}

// ---------------- layer-1 GEMM: H = nsrc * (X @ W1)  [N,256]x[256,128] ----------------
// 256 threads = 8 waves; block owns a 16-row strip; wave w owns cols [16w,16w+16).
// A fragment (f32 16x4): lanes 0-15 K={k0,k0+1}, lanes 16-31 K={k0+2,k0+3}, M=lane&15.
__global__ __launch_bounds__(256)
void gcn_gemm1_wmma(const float* __restrict__ X, const float* __restrict__ nsrc,
                    const float* __restrict__ W, float* __restrict__ H, int nrows)
{
    __shared__ float sA[16][IN_F + 4];   // row pitch 1040B: 16B aligned, banks spread
    const int m0   = blockIdx.x * 16;
    const int lane = threadIdx.x & 31;
    const int wave = threadIdx.x >> 5;

    // Stage the raw 16x256 strip: 1024 x 16B async copies, 4 per thread.
    for (int i = threadIdx.x; i < 16 * (IN_F / 4); i += 256) {
        int r = i >> 6;                 // chunk row (64 chunks per row)
        int c = (i & 63) * 4;           // starting float within row
        int row = m0 + r;
        if (row < nrows) {
            async_g2l_b128(&sA[r][c], X + (size_t)row * IN_F + c);
        } else {
            sA[r][c] = 0.0f; sA[r][c+1] = 0.0f; sA[r][c+2] = 0.0f; sA[r][c+3] = 0.0f;
        }
    }
    wait_async0();
    __syncthreads();

    const int hi   = lane >> 4;         // which K-half this lane supplies
    const int lidx = lane & 15;
    const int col  = wave * 16 + lidx;

    v8f acc = {};
#pragma unroll 4
    for (int k0 = 0; k0 < IN_F; k0 += 4) {
        int ka = k0 + hi * 2;
        v2f a = *(const v2f*)&sA[lidx][ka];              // ds_load_b64
        v2f b;
        b.x = W[(size_t)ka * H_F + col];
        b.y = W[(size_t)(ka + 1) * H_F + col];
        acc = __builtin_amdgcn_wmma_f32_16x16x4_f32(
            false, a, false, b, (short)0, acc, false, false);
    }

    // Epilogue: fold norm_src row scaling here; uniform full-tile fast path.
    if (m0 + 16 <= nrows) {
#pragma unroll
        for (int v = 0; v < 8; ++v) {
            int row = m0 + v + hi * 8;                   // C layout: M = v + 8*half
            H[(size_t)row * H_F + col] = acc[v] * nsrc[row];
        }
    } else {
#pragma unroll
        for (int v = 0; v < 8; ++v) {
            int row = m0 + v + hi * 8;
            if (row < nrows) H[(size_t)row * H_F + col] = acc[v] * nsrc[row];
        }
    }
}

// ---------------- layer-2 GEMM: H2 = nsrc * (Z @ W2p)  [N,128]x[128,48] ----------------
__global__ __launch_bounds__(96)
void gcn_gemm2_wmma(const float* __restrict__ Z, const float* __restrict__ nsrc,
                    const float* __restrict__ Wp, float* __restrict__ H2, int nrows)
{
    __shared__ float sA[16][H_F + 4];    // row pitch 528B: 16B aligned
    const int m0   = blockIdx.x * 16;
    const int lane = threadIdx.x & 31;
    const int wave = threadIdx.x >> 5;   // 3 waves -> cols 0..47

    for (int i = threadIdx.x; i < 16 * (H_F / 4); i += 96) {
        int r = i >> 5;                  // 32 chunks per row
        int c = (i & 31) * 4;
        int row = m0 + r;
        if (row < nrows) {
            async_g2l_b128(&sA[r][c], Z + (size_t)row * H_F + c);
        } else {
            sA[r][c] = 0.0f; sA[r][c+1] = 0.0f; sA[r][c+2] = 0.0f; sA[r][c+3] = 0.0f;
        }
    }
    wait_async0();
    __syncthreads();

    const int hi   = lane >> 4;
    const int lidx = lane & 15;
    const int col  = wave * 16 + lidx;

    v8f acc = {};
#pragma unroll 4
    for (int k0 = 0; k0 < H_F; k0 += 4) {
        int ka = k0 + hi * 2;
        v2f a = *(const v2f*)&sA[lidx][ka];
        v2f b;
        b.x = Wp[(size_t)ka * N_PAD + col];
        b.y = Wp[(size_t)(ka + 1) * N_PAD + col];
        acc = __builtin_amdgcn_wmma_f32_16x16x4_f32(
            false, a, false, b, (short)0, acc, false, false);
    }

    if (col < N_CLS) {
        if (m0 + 16 <= nrows) {
#pragma unroll
            for (int v = 0; v < 8; ++v) {
                int row = m0 + v + hi * 8;
                H2[(size_t)row * N_CLS + col] = acc[v] * nsrc[row];
            }
        } else {
#pragma unroll
            for (int v = 0; v < 8; ++v) {
                int row = m0 + v + hi * 8;
                if (row < nrows) H2[(size_t)row * N_CLS + col] = acc[v] * nsrc[row];
            }
        }
    }
}

// ---------------- edge scatter, layer 1: A[dst] += H[src]  (128 f32/edge) ----------------
// One wave per edge: lanes do one float4 gather (coalesced 512B/edge) + 4 f32 atomics.
__global__ __launch_bounds__(256)
void gcn_scatter1(const float* __restrict__ H, const int* __restrict__ src,
                  const int* __restrict__ dst, float* __restrict__ A, int E)
{
    int gwave = (int)((blockIdx.x * (unsigned)blockDim.x + threadIdx.x) >> 5);
    int lane  = threadIdx.x & 31;
    int nwave = (int)((gridDim.x * (unsigned)blockDim.x) >> 5);
    for (int e = gwave; e < E; e += nwave) {
        int s = src[e], d = dst[e];                       // wave-uniform -> s_load
        float4 hv = ((const float4*)(H + (size_t)s * H_F))[lane];
        float* ap = A + (size_t)d * H_F + lane * 4;
        atomicAdd(ap + 0, hv.x);
        atomicAdd(ap + 1, hv.y);
        atomicAdd(ap + 2, hv.z);
        atomicAdd(ap + 3, hv.w);
    }
}

// ---------------- bias + ReLU (layer 1), writes Z over the H buffer ----------------
__global__ void gcn_relu_bias(const float* __restrict__ A, const float* __restrict__ ndst,
                              const float* __restrict__ b, float* __restrict__ Z, int n)
{
    int i = blockIdx.x * blockDim.x + threadIdx.x;
    if (i < n) {
        int row = i >> 7, c = i & (H_F - 1);
        Z[i] = fmaxf(A[i] * ndst[row] + b[c], 0.0f);
    }
}

// ---------------- edge scatter, layer 2: out[dst] += H2[src] (40 f32/edge) ----------------
__global__ void gcn_scatter2(const float* __restrict__ H2, const int* __restrict__ src,
                             const int* __restrict__ dst, float* __restrict__ out, int E)
{
    long long i = (long long)blockIdx.x * blockDim.x + threadIdx.x;
    long long tot = (long long)E * N_CLS;
    long long stride = (long long)gridDim.x * blockDim.x;
    for (; i < tot; i += stride) {
        int e = (int)(i / N_CLS);
        int c = (int)(i - (long long)e * N_CLS);
        atomicAdd(&out[(size_t)dst[e] * N_CLS + c], H2[(size_t)src[e] * N_CLS + c]);
    }
}

// ---------------- final scale + bias (in place on d_out) ----------------
__global__ void gcn_out_fin(float* __restrict__ out, const float* __restrict__ ndst,
                            const float* __restrict__ b, int n)
{
    int i = blockIdx.x * blockDim.x + threadIdx.x;
    if (i < n) {
        int row = i / N_CLS;
        int c = i - row * N_CLS;
        out[i] = out[i] * ndst[row] + b[c];
    }
}

extern "C" void kernel_launch(void* const* d_in, const int* in_sizes, int n_in,
                              void* d_out, int out_size, void* d_ws, size_t ws_size,
                              hipStream_t stream)
{
    const float* X   = (const float*)d_in[0];
    const int*   src = (const int*)d_in[1];
    const int*   dst = (const int*)d_in[2];
    const float* W1  = (const float*)d_in[3];
    const float* b1  = (const float*)d_in[4];
    const float* W2  = (const float*)d_in[5];
    const float* b2  = (const float*)d_in[6];
    float* out = (float*)d_out;

    const int Nn = in_sizes[0] / IN_F;   // 100000
    const int E  = in_sizes[1];          // 1600000

    // workspace layout (~103 MB): norms | H [N,128] | A1 [N,128] (reused as H2) | W2pad
    float* norm_src = (float*)d_ws;
    float* norm_dst = norm_src + Nn;
    float* Hbuf     = norm_dst + Nn;
    float* A1       = Hbuf + (size_t)Nn * H_F;
    float* W2p      = A1 + (size_t)Nn * H_F;

    hipMemsetAsync(norm_src, 0, sizeof(float) * 2 * (size_t)Nn, stream);
    hipMemsetAsync(A1, 0, sizeof(float) * (size_t)Nn * H_F, stream);
    hipMemsetAsync(out, 0, sizeof(float) * (size_t)out_size, stream);

    gcn_degree_kernel<<<2048, 256, 0, stream>>>(src, dst, norm_src, norm_dst, E);
    gcn_norm_kernel<<<(Nn + 255) / 256, 256, 0, stream>>>(norm_src, norm_dst, Nn);
    gcn_padw2<<<(H_F * N_PAD + 255) / 256, 256, 0, stream>>>(W2, W2p);

    const int mblocks = (Nn + 15) / 16;  // 6250, exact tiles (100000 = 6250*16)
    gcn_gemm1_wmma<<<mblocks, 256, 0, stream>>>(X, norm_src, W1, Hbuf, Nn);

    const int sblocks = (E + 7) / 8;     // 8 waves/block, one edge per wave
    gcn_scatter1<<<sblocks, 256, 0, stream>>>(Hbuf, src, dst, A1, E);

    gcn_relu_bias<<<(int)(((size_t)Nn * H_F + 255) / 256), 256, 0, stream>>>(
        A1, norm_dst, b1, Hbuf, Nn * H_F);

    gcn_gemm2_wmma<<<mblocks, 96, 0, stream>>>(Hbuf, norm_src, W2p, A1 /*H2*/, Nn);

    long long tot2 = (long long)E * N_CLS;
    gcn_scatter2<<<(int)((tot2 + 255) / 256), 256, 0, stream>>>(A1, src, dst, out, E);

    gcn_out_fin<<<(int)(((size_t)Nn * N_CLS + 255) / 256), 256, 0, stream>>>(
        out, norm_dst, b2, Nn * N_CLS);
}